// GlobalHeteroEncoder_88441966559849
// MI455X (gfx1250) — compile-verified
//
#include <hip/hip_runtime.h>
#include <math.h>

#define NNODES 50000
#define NEDGES 800000
#define FDIM   128
#define NHEADS 4
#define HD     32
#define NEG_SLOPE 0.2f

typedef __attribute__((ext_vector_type(2))) float v2f;
typedef __attribute__((ext_vector_type(8))) float v8f;

// ---------------------------------------------------------------------------
// h = X @ W   (X: [n,128] row-major, W: [128,128] row-major, h: [n,128])
// One wave computes one 16x16 output tile with V_WMMA_F32_16X16X4_F32,
// looping K in steps of 4 (32 WMMA ops per tile). 8 waves/block cover the
// full 128-wide output strip of 16 rows. n must be a multiple of 16 (50000 is).
// ---------------------------------------------------------------------------
__global__ __launch_bounds__(256) void gemm_f32_wmma(const float* __restrict__ X,
                                                     const float* __restrict__ W,
                                                     float* __restrict__ Hout) {
  const int lane = threadIdx.x & 31;
  const int wave = threadIdx.x >> 5;
  const int row0 = blockIdx.x << 4;   // 16-row strip
  const int col0 = wave << 4;         // 16-col tile within the 128 cols
  const int half = lane >> 4;         // 0: lanes 0-15, 1: lanes 16-31
  const int l15  = lane & 15;
  const int arow = row0 + l15;        // A: M = lane&15 in both halves
  const int bcol = col0 + l15;        // B: N = lane&15 in both halves

  v8f acc = {0.f, 0.f, 0.f, 0.f, 0.f, 0.f, 0.f, 0.f};
#pragma unroll
  for (int k0 = 0; k0 < FDIM; k0 += 4) {
    const int ka = k0 + (half << 1);  // lane-half selects K pair {0,1} or {2,3}
    v2f a, b;
    // A 16x4 f32: VGPR v holds K = 2*half + v for row M = lane&15
    a.x = X[arow * FDIM + ka];
    a.y = X[arow * FDIM + ka + 1];
    // B 4x16 f32: VGPR v holds K = 2*half + v for col N = lane&15
    b.x = W[ka * FDIM + bcol];
    b.y = W[(ka + 1) * FDIM + bcol];
    acc = __builtin_amdgcn_wmma_f32_16x16x4_f32(
        /*neg_a=*/false, a, /*neg_b=*/false, b,
        /*c_mod=*/(short)0, acc, /*reuse_a=*/false, /*reuse_b=*/false);
  }
  // C/D 16x16 f32: VGPR r -> row r (lanes 0-15) / row r+8 (lanes 16-31)
#pragma unroll
  for (int r = 0; r < 8; ++r) {
    Hout[(row0 + (half << 3) + r) * FDIM + col0 + l15] = acc[r];
  }
}

// ---------------------------------------------------------------------------
// alpha_s[n,h] = dot(h[n,h,:], a_src[h,:]);  alpha_d likewise.
// One thread per (node, head): reads 32 contiguous floats.
// ---------------------------------------------------------------------------
__global__ void alpha_kernel(const float* __restrict__ h,
                             const float* __restrict__ a_src,
                             const float* __restrict__ a_dst,
                             float* __restrict__ alphaS,
                             float* __restrict__ alphaD) {
  const int idx = blockIdx.x * blockDim.x + threadIdx.x;
  if (idx >= NNODES * NHEADS) return;
  const int n = idx >> 2;
  const int head = idx & 3;
  const float* hp = h + (size_t)n * FDIM + head * HD;
  const float* sp = a_src + head * HD;
  const float* dp = a_dst + head * HD;
  float s = 0.f, d = 0.f;
#pragma unroll
  for (int i = 0; i < HD; ++i) {
    const float v = hp[i];
    s += v * sp[i];
    d += v * dp[i];
  }
  alphaS[idx] = s;
  alphaD[idx] = d;
}

// order-preserving float <-> uint mapping for atomic max on floats
__device__ __forceinline__ unsigned f2key(float f) {
  unsigned b = __float_as_uint(f);
  return (b & 0x80000000u) ? ~b : (b | 0x80000000u);
}
__device__ __forceinline__ float key2f(unsigned k) {
  return (k & 0x80000000u) ? __uint_as_float(k ^ 0x80000000u) : __uint_as_float(~k);
}
__device__ __forceinline__ float leaky(float x) {
  return x > 0.f ? x : NEG_SLOPE * x;
}

// per (edge, head): logits = leakyrelu(aS[src]+aD[dst]); atomic segment max on dst
__global__ void edge_max_kernel(const float* __restrict__ alphaS,
                                const float* __restrict__ alphaD,
                                const long long* __restrict__ src,
                                const long long* __restrict__ dst,
                                unsigned* __restrict__ segmax) {
  const int idx = blockIdx.x * blockDim.x + threadIdx.x;
  if (idx >= NEDGES * NHEADS) return;
  const int e = idx >> 2;
  const int head = idx & 3;
  const int s = (int)src[e];
  const int d = (int)dst[e];
  const float lg = leaky(alphaS[s * NHEADS + head] + alphaD[d * NHEADS + head]);
  atomicMax(&segmax[d * NHEADS + head], f2key(lg));
}

// per (edge, head): ex = exp(logit - segmax[dst]); atomic add into denom[dst]
__global__ void edge_denom_kernel(const float* __restrict__ alphaS,
                                  const float* __restrict__ alphaD,
                                  const long long* __restrict__ src,
                                  const long long* __restrict__ dst,
                                  const unsigned* __restrict__ segmax,
                                  float* __restrict__ denom) {
  const int idx = blockIdx.x * blockDim.x + threadIdx.x;
  if (idx >= NEDGES * NHEADS) return;
  const int e = idx >> 2;
  const int head = idx & 3;
  const int s = (int)src[e];
  const int d = (int)dst[e];
  const float lg = leaky(alphaS[s * NHEADS + head] + alphaD[d * NHEADS + head]);
  const unsigned k = segmax[d * NHEADS + head];
  const float mx = (k == 0u) ? 0.f : key2f(k);
  atomicAdd(&denom[d * NHEADS + head], expf(lg - mx));
}

// one wave per edge: recompute att per head, scatter att*h[src] into agg[dst]
__global__ __launch_bounds__(256) void aggregate_kernel(const float* __restrict__ h,
                                                        const float* __restrict__ alphaS,
                                                        const float* __restrict__ alphaD,
                                                        const long long* __restrict__ src,
                                                        const long long* __restrict__ dst,
                                                        const unsigned* __restrict__ segmax,
                                                        const float* __restrict__ denom,
                                                        float* __restrict__ agg) {
  const int e = blockIdx.x * (blockDim.x >> 5) + (threadIdx.x >> 5);
  if (e >= NEDGES) return;
  const int lane = threadIdx.x & 31;
  const int s = (int)src[e];
  const int d = (int)dst[e];
  float att[NHEADS];
#pragma unroll
  for (int head = 0; head < NHEADS; ++head) {
    const float lg = leaky(alphaS[s * NHEADS + head] + alphaD[d * NHEADS + head]);
    const unsigned k = segmax[d * NHEADS + head];
    const float mx = (k == 0u) ? 0.f : key2f(k);
    const float den = fmaxf(denom[d * NHEADS + head], 1e-16f);
    att[head] = expf(lg - mx) / den;
  }
#pragma unroll
  for (int head = 0; head < NHEADS; ++head) {
    const int c = head * HD + lane;
    atomicAdd(&agg[(size_t)d * FDIM + c], att[head] * h[(size_t)s * FDIM + c]);
  }
}

// out = ELU(agg + bias)
__global__ void bias_elu_kernel(const float* __restrict__ agg,
                                const float* __restrict__ bias,
                                float* __restrict__ out) {
  const int idx = blockIdx.x * blockDim.x + threadIdx.x;
  if (idx >= NNODES * FDIM) return;
  const float v = agg[idx] + bias[idx & (FDIM - 1)];
  out[idx] = v > 0.f ? v : expm1f(v);
}

// ---------------------------------------------------------------------------
extern "C" void kernel_launch(void* const* d_in, const int* in_sizes, int n_in,
                              void* d_out, int out_size, void* d_ws, size_t ws_size,
                              hipStream_t stream) {
  (void)in_sizes; (void)n_in; (void)out_size; (void)ws_size;
  const float*     X    = (const float*)d_in[0];
  const long long* ei   = (const long long*)d_in[1];   // int64 [2, E]
  const float*     W0   = (const float*)d_in[2];
  const float*     aS0  = (const float*)d_in[3];
  const float*     aD0  = (const float*)d_in[4];
  const float*     b0   = (const float*)d_in[5];
  const float*     W1   = (const float*)d_in[6];
  const float*     aS1  = (const float*)d_in[7];
  const float*     aD1  = (const float*)d_in[8];
  const float*     b1   = (const float*)d_in[9];
  const long long* src  = ei;
  const long long* dst  = ei + NEDGES;

  char* ws = (char*)d_ws;
  size_t off = 0;
  const size_t featBytes = (size_t)NNODES * FDIM * sizeof(float);
  const size_t nhBytes   = (size_t)NNODES * NHEADS * sizeof(float);
  float*    bufB   = (float*)(ws + off);   off += featBytes;  // transformed features h
  float*    bufA   = (float*)(ws + off);   off += featBytes;  // aggregation / activations
  float*    alphaS = (float*)(ws + off);   off += nhBytes;
  float*    alphaD = (float*)(ws + off);   off += nhBytes;
  unsigned* segmax = (unsigned*)(ws + off); off += nhBytes;
  float*    denom  = (float*)(ws + off);   off += nhBytes;

  const int gemmBlocks  = NNODES / 16;                       // 3125
  const int alphaBlocks = (NNODES * NHEADS + 255) / 256;     // 782
  const int edgeBlocks  = (NEDGES * NHEADS + 255) / 256;     // 12500
  const int aggBlocks   = (NEDGES + 7) / 8;                  // 100000
  const int beBlocks    = (NNODES * FDIM + 255) / 256;       // 25000

  // ---------------- Layer 1 ----------------
  hipMemsetAsync(bufA,   0, featBytes, stream);
  hipMemsetAsync(segmax, 0, nhBytes,   stream);
  hipMemsetAsync(denom,  0, nhBytes,   stream);
  gemm_f32_wmma<<<gemmBlocks, 256, 0, stream>>>(X, W0, bufB);
  alpha_kernel<<<alphaBlocks, 256, 0, stream>>>(bufB, aS0, aD0, alphaS, alphaD);
  edge_max_kernel<<<edgeBlocks, 256, 0, stream>>>(alphaS, alphaD, src, dst, segmax);
  edge_denom_kernel<<<edgeBlocks, 256, 0, stream>>>(alphaS, alphaD, src, dst, segmax, denom);
  aggregate_kernel<<<aggBlocks, 256, 0, stream>>>(bufB, alphaS, alphaD, src, dst,
                                                  segmax, denom, bufA);
  bias_elu_kernel<<<beBlocks, 256, 0, stream>>>(bufA, b0, bufA);  // bufA = H (layer-2 input)

  // ---------------- Layer 2 ----------------
  gemm_f32_wmma<<<gemmBlocks, 256, 0, stream>>>(bufA, W1, bufB);  // bufA free after this
  hipMemsetAsync(bufA,   0, featBytes, stream);
  hipMemsetAsync(segmax, 0, nhBytes,   stream);
  hipMemsetAsync(denom,  0, nhBytes,   stream);
  alpha_kernel<<<alphaBlocks, 256, 0, stream>>>(bufB, aS1, aD1, alphaS, alphaD);
  edge_max_kernel<<<edgeBlocks, 256, 0, stream>>>(alphaS, alphaD, src, dst, segmax);
  edge_denom_kernel<<<edgeBlocks, 256, 0, stream>>>(alphaS, alphaD, src, dst, segmax, denom);
  aggregate_kernel<<<aggBlocks, 256, 0, stream>>>(bufB, alphaS, alphaD, src, dst,
                                                  segmax, denom, bufA);
  bias_elu_kernel<<<beBlocks, 256, 0, stream>>>(bufA, b1, (float*)d_out);
}